// Sinkhorn_10359461117952
// MI455X (gfx1250) — compile-verified
//
#include <hip/hip_runtime.h>
#include <hip/hip_bf16.h>
#include <stdint.h>

// Sinkhorn (log domain), B=8, N=1024, 20 iters, tau=1.
// 128 persistent workgroups (8 matrices x 16 row-slabs of 64 rows). Each slab
// (64x1024 f32 = 256KB) lives in LDS for all 20 iterations (320KB LDS/WGP).
// Row LSE is slab-local; column LSE exchanges only per-column (max,sumexp)
// partials through L2 with one grid barrier per iteration (double-buffered).
// Global<->LDS slab movement uses the CDNA5 Tensor Data Mover in both
// directions (tensor_load_to_lds / tensor_store_from_lds).

#define N        1024
#define BATCH    8
#define ROWS_WG  64
#define WGS_MAT  16
#define NWG      (BATCH * WGS_MAT)   // 128
#define NITERS   20
#define FLT_NEG  (-3.402823466e38f)

typedef unsigned int u32x4 __attribute__((ext_vector_type(4)));
typedef int          i32x8 __attribute__((ext_vector_type(8)));
typedef int          i32x4 __attribute__((ext_vector_type(4)));

__device__ __forceinline__ float wave_max(float v) {
#pragma unroll
  for (int off = 16; off >= 1; off >>= 1)
    v = fmaxf(v, __shfl_xor(v, off, 32));
  return v;
}
__device__ __forceinline__ float wave_sum(float v) {
#pragma unroll
  for (int off = 16; off >= 1; off >>= 1)
    v += __shfl_xor(v, off, 32);
  return v;
}

// Build a 2D TDM descriptor for a 64x1024 f32 tile of a 1024-wide tensor.
__device__ __forceinline__ void tdm_desc(u32x4& g0, i32x8& g1,
                                         unsigned lds_off,
                                         unsigned long long ga) {
  g0[0] = 1u;                                   // count=1, user mode
  g0[1] = lds_off;                              // lds_addr (bytes)
  g0[2] = (unsigned)(ga & 0xFFFFFFFFull);       // global_addr[31:0]
  g0[3] = (unsigned)((ga >> 32) & 0x1FFFFFFull) // global_addr[56:32]
        | (2u << 30);                           // type=2 ("image")
  g1[0] = (2 << 16);                            // data_size = 2 -> 4 bytes
  g1[1] = (int)((N & 0xFFFF) << 16);            // tensor_dim0[15:0] @ [63:48]
  g1[2] = (int)((N >> 16) | ((N & 0xFFFF) << 16)); // dim0 hi | tensor_dim1 lo
  g1[3] = (int)((N >> 16) | (N << 16));            // dim1 hi | tile_dim0=1024
  g1[4] = ROWS_WG;                              // tile_dim1=64, tile_dim2=0
  g1[5] = N;                                    // tensor_dim0_stride = 1024
  g1[6] = 0;
  g1[7] = 0;
}

// Device-wide split barrier over NWG resident workgroups.
__device__ __forceinline__ void grid_sync(unsigned* cnt, volatile unsigned* gen) {
  __threadfence();
  __syncthreads();
  if (threadIdx.x == 0) {
    unsigned g = *gen;
    if (atomicAdd(cnt, 1u) == (unsigned)(NWG - 1)) {
      atomicExch(cnt, 0u);
      __threadfence();
      atomicAdd((unsigned*)gen, 1u);
    } else {
      while (*gen == g) { __builtin_amdgcn_s_sleep(2); }
    }
  }
  __syncthreads();
  __threadfence();
}

extern "C" __global__ __launch_bounds__(1024)
void sinkhorn_persistent(const float* __restrict__ cost,
                         float* __restrict__ out,
                         float2* __restrict__ partials,  // [2][8][16][1024]
                         float* __restrict__ costp,      // [128]
                         unsigned* __restrict__ sync_cnt,
                         unsigned* __restrict__ sync_gen)
{
  extern __shared__ float tile[];   // 64 x 1024 f32 = 256 KB
  __shared__ float red[32];

  const int tid  = threadIdx.x;            // 0..1023
  const int lane = tid & 31;
  const int wave = tid >> 5;               // 0..31
  const int mat  = blockIdx.x / WGS_MAT;   // 0..7
  const int wgm  = blockIdx.x % WGS_MAT;   // 0..15
  const int row0 = wgm * ROWS_WG;
  const float* cmat = cost + (size_t)mat * N * N;
  const unsigned lds_off = (unsigned)(uintptr_t)(void*)&tile[0];

  // ---------------- TDM load: 64x1024 f32 tile, global -> LDS -------------
  if (wave == 0) {
    u32x4 g0; i32x8 g1;
    tdm_desc(g0, g1, lds_off,
             (unsigned long long)(uintptr_t)(cmat + (size_t)row0 * N));
    i32x4 g2 = {0, 0, 0, 0};
    i32x4 g3 = {0, 0, 0, 0};
    i32x8 g4 = {0, 0, 0, 0, 0, 0, 0, 0};
    __builtin_amdgcn_tensor_load_to_lds(g0, g1, g2, g3, g4, 0);
    __builtin_amdgcn_s_wait_tensorcnt(0);
  }
  __syncthreads();

  // ---------------- 20 Sinkhorn iterations --------------------------------
  // Z = -C/tau (tau=1): negation folded into iteration 0's row pass.
  for (int it = 0; it < NITERS; ++it) {
    // ---- Row normalize (slab-local). Wave w handles rows w and w+32. ----
#pragma unroll
    for (int rr = 0; rr < 2; ++rr) {
      const int r = wave + rr * 32;
      float vals[32];
      float m = FLT_NEG;
#pragma unroll
      for (int k = 0; k < 32; ++k) {
        float t = tile[r * N + lane + 32 * k];
        vals[k] = (it == 0) ? -t : t;           // uniform-predicated negate
        m = fmaxf(m, vals[k]);
      }
      m = wave_max(m);
      float s = 0.f;
#pragma unroll
      for (int k = 0; k < 32; ++k) s += __expf(vals[k] - m);
      s = wave_sum(s);
      const float lse = m + __logf(s);
#pragma unroll
      for (int k = 0; k < 32; ++k)
        tile[r * N + lane + 32 * k] = vals[k] - lse;
    }
    __syncthreads();

    // ---- Column partials (local): thread tid reduces column tid. ----
    float cm = FLT_NEG;
#pragma unroll 8
    for (int r = 0; r < ROWS_WG; ++r)
      cm = fmaxf(cm, tile[r * N + tid]);
    float cs = 0.f;
#pragma unroll 8
    for (int r = 0; r < ROWS_WG; ++r)
      cs += __expf(tile[r * N + tid] - cm);
    const int buf = it & 1;
    partials[(((size_t)buf * BATCH + mat) * WGS_MAT + wgm) * N + tid] =
        make_float2(cm, cs);

    grid_sync(sync_cnt, sync_gen);

    // ---- Column combine: gather 16 partials, subtract column LSE. ----
    float2 pk[WGS_MAT];
    float M = FLT_NEG;
#pragma unroll
    for (int k = 0; k < WGS_MAT; ++k) {
      pk[k] = partials[(((size_t)buf * BATCH + mat) * WGS_MAT + k) * N + tid];
      M = fmaxf(M, pk[k].x);
    }
    float S = 0.f;
#pragma unroll
    for (int k = 0; k < WGS_MAT; ++k)
      S += pk[k].y * __expf(pk[k].x - M);
    const float lse = M + __logf(S);
#pragma unroll 8
    for (int r = 0; r < ROWS_WG; ++r)
      tile[r * N + tid] -= lse;
    __syncthreads();
  }

  // ------- P = exp(Z) into LDS, cost partial = sum(P * C) in regs ---------
  float cacc = 0.f;
#pragma unroll
  for (int rr = 0; rr < 2; ++rr) {
    const int r = wave + rr * 32;
    const int grow = row0 + r;
    const float* crow = cmat + (size_t)grow * N;
#pragma unroll 4
    for (int k = 0; k < 32; ++k) {
      const int c = lane + 32 * k;
      const float p = __expf(tile[r * N + c]);
      tile[r * N + c] = p;          // tile now holds P
      cacc += p * crow[c];
    }
  }
  __syncthreads();

  // ---------------- TDM store: LDS tile -> out (P slab) -------------------
  if (wave == 0) {
    u32x4 g0; i32x8 g1;
    tdm_desc(g0, g1, lds_off,
             (unsigned long long)(uintptr_t)
                 (out + ((size_t)mat * N + row0) * N));
    i32x4 g2 = {0, 0, 0, 0};
    i32x4 g3 = {0, 0, 0, 0};
    i32x8 g4 = {0, 0, 0, 0, 0, 0, 0, 0};
    __builtin_amdgcn_tensor_store_from_lds(g0, g1, g2, g3, g4, 0);
    // no explicit wait: s_endpgm's implicit wait-idle drains TENSORcnt,
    // and LDS stays allocated until the workgroup fully retires.
  }

  // ---------------- block-level cost reduction (overlaps TDM store) -------
  cacc = wave_sum(cacc);
  if (lane == 0) red[wave] = cacc;
  __syncthreads();
  if (wave == 0) {
    float v = red[lane];
    v = wave_sum(v);
    if (lane == 0) costp[blockIdx.x] = v;
  }
}

extern "C" __global__ void cost_reduce(const float* __restrict__ costp,
                                       float* __restrict__ out)
{
  const int b = blockIdx.x;          // 0..7
  const int lane = threadIdx.x;      // 0..31
  float v = (lane < WGS_MAT) ? costp[b * WGS_MAT + lane] : 0.f;
  v = wave_sum(v);
  if (lane == 0) out[(size_t)BATCH * N * N + b] = v;
}

extern "C" void kernel_launch(void* const* d_in, const int* in_sizes, int n_in,
                              void* d_out, int out_size, void* d_ws, size_t ws_size,
                              hipStream_t stream) {
  const float* cost = (const float*)d_in[0];
  float* out = (float*)d_out;

  char* ws = (char*)d_ws;
  unsigned* sync_cnt = (unsigned*)(ws + 0);
  unsigned* sync_gen = (unsigned*)(ws + 64);
  float2*   partials = (float2*)(ws + 256);                       // 2 MB
  float*    costp    = (float*)(ws + 256 +
                        (size_t)2 * BATCH * WGS_MAT * N * sizeof(float2));

  // Reset barrier state every call (d_ws is poisoned once, never re-poisoned).
  (void)hipMemsetAsync(d_ws, 0, 256, stream);

  const size_t lds_bytes = (size_t)ROWS_WG * N * sizeof(float);   // 256 KB
  sinkhorn_persistent<<<dim3(NWG), dim3(1024), lds_bytes, stream>>>(
      cost, out, partials, costp, sync_cnt, sync_gen);
  cost_reduce<<<dim3(BATCH), dim3(32), 0, stream>>>(costp, out);
}